// Transformer_59794534694944
// MI455X (gfx1250) — compile-verified
//
#include <hip/hip_runtime.h>
#include <hip/hip_bf16.h>
#include <math.h>

#define L   6
#define H   16
#define D   1024
#define V   32000
#define CTX 1024
#define BB  2
#define HD  64
#define NT  (BB * CTX)   // 2048 tokens

typedef float v2f __attribute__((ext_vector_type(2)));
typedef float v8f __attribute__((ext_vector_type(8)));

// LDS tile row pads chosen so async B128 LDS destinations are 16B-aligned:
// A rows: 20 floats (80 B = 5*16), B rows: 68 floats (272 B = 17*16).
#define APAD 20
#define BPAD 68

// ---------------------------------------------------------------- embedding
__global__ void embed_kernel(const int* __restrict__ x,
                             const float* __restrict__ tok,
                             const float* __restrict__ pos,
                             float* __restrict__ h) {
  int t = blockIdx.x;            // token 0..NT-1
  int c = t % CTX;
  int id = x[t];
  const float* te = tok + (long)id * D;
  const float* pe = pos + (long)c * D;
  float* out = h + (long)t * D;
  for (int d = threadIdx.x; d < D; d += blockDim.x)
    out[d] = te[d] + pe[d];
}

// ---------------------------------------------------------------- layernorm
__global__ __launch_bounds__(256)
void ln_kernel(const float* __restrict__ in, const float* __restrict__ w,
               const float* __restrict__ b, float* __restrict__ out) {
  __shared__ float red[256];
  int t = blockIdx.x;
  const float* row = in + (long)t * D;
  float s = 0.f;
  for (int d = threadIdx.x; d < D; d += 256) s += row[d];
  red[threadIdx.x] = s; __syncthreads();
  for (int o = 128; o > 0; o >>= 1) {
    if (threadIdx.x < o) red[threadIdx.x] += red[threadIdx.x + o];
    __syncthreads();
  }
  float mean = red[0] * (1.0f / D); __syncthreads();
  float vs = 0.f;
  for (int d = threadIdx.x; d < D; d += 256) { float df = row[d] - mean; vs += df * df; }
  red[threadIdx.x] = vs; __syncthreads();
  for (int o = 128; o > 0; o >>= 1) {
    if (threadIdx.x < o) red[threadIdx.x] += red[threadIdx.x + o];
    __syncthreads();
  }
  float rstd = rsqrtf(red[0] * (1.0f / D) + 1e-5f);
  float* orow = out + (long)t * D;
  for (int d = threadIdx.x; d < D; d += 256)
    orow[d] = (row[d] - mean) * rstd * w[d] + b[d];
}

// ---------------------------------------------------------------- GELU (tanh)
__device__ __forceinline__ float gelu_f(float x) {
  const float c = 0.7978845608028654f;     // sqrt(2/pi)
  return 0.5f * x * (1.0f + tanhf(c * (x + 0.044715f * x * x * x)));
}

// ----------------------------------------------------- async global->LDS copy
__device__ __forceinline__ void async_copy_b128(unsigned lds_off, const float* gptr) {
  asm volatile("global_load_async_to_lds_b128 %0, %1, off"
               :: "v"(lds_off), "v"(gptr) : "memory");
}
__device__ __forceinline__ void async_copy_b32(unsigned lds_off, const float* gptr) {
  asm volatile("global_load_async_to_lds_b32 %0, %1, off"
               :: "v"(lds_off), "v"(gptr) : "memory");
}
template <int N>
__device__ __forceinline__ void wait_asynccnt() {
  asm volatile("s_wait_asynccnt %0" :: "i"(N) : "memory");
}
__device__ __forceinline__ unsigned lds_off_of(const void* p) {
  return (unsigned)(uintptr_t)p;   // generic LDS addr: low 32 bits = LDS offset
}

// ---------------------------------------------------------------- batched WMMA GEMM
// C[z] = act( alpha * A[z] (@ or @^T) B[z] + bias ) + residual
// Block: 256 threads (8 waves). Block tile 128(M) x 64(N), K-step 16.
// Wave tile: 32 x 32 (four 16x16 f32 accumulators), V_WMMA_F32_16X16X4_F32.
// Double-buffered LDS tiles filled with GLOBAL_LOAD_ASYNC_TO_LDS (ASYNCcnt).
// C base offset per z: (z % groupC)*strideC1 + (z / groupC)*strideC2.
template <int TRANSB>
__global__ __launch_bounds__(256)
void gemm_wmma_kernel(const float* __restrict__ A, const float* __restrict__ B,
                      const float* __restrict__ bias, const float* __restrict__ residual,
                      float* __restrict__ Cmat,
                      int M, int N, int K, int lda, int ldb, int ldc,
                      long strideA, long strideB,
                      int groupC, long strideC1, long strideC2,
                      float alpha, int act_gelu) {
  __shared__ float As[2][128][APAD];
  __shared__ float Bs[2][16][BPAD];

  int z = blockIdx.z;
  const float* Ab = A + (long)z * strideA;
  const float* Bb = B + (long)z * strideB;
  float* Cb = Cmat + (long)(z % groupC) * strideC1 + (long)(z / groupC) * strideC2;

  int n0 = blockIdx.x * 64;
  int m0 = blockIdx.y * 128;
  int tid  = threadIdx.x;
  int wid  = tid >> 5;
  int lane = tid & 31;
  int wm = wid & 3;        // 0..3: wave row group (32 rows each)
  int wn = wid >> 2;       // 0..1: wave col group (32 cols each)
  int lrow = lane & 15;
  int sel  = lane >> 4;    // half-wave select

  // per-thread async-copy assignments
  int arow = tid >> 1;            // 0..127
  int acol = (tid & 1) * 8;       // 0 or 8  (two B128s: +0, +4 floats)
  int brow = tid >> 4;            // 0..15   (TRANSB==0)
  int bcol = (tid & 15) * 4;      // 0..60
  int tn   = tid & 63;            // 0..63   (TRANSB==1)
  int tkk  = (tid >> 6) * 4;      // 0,4,8,12

  // async ops issued per thread per tile
  const int NPF = TRANSB ? 6 : 3;

  auto issue_tile = [&](int k0, int buf) {
    const float* ap = Ab + (long)(m0 + arow) * lda + (k0 + acol);
    unsigned al = lds_off_of(&As[buf][arow][acol]);
    async_copy_b128(al, ap);
    async_copy_b128(al + 16, ap + 4);
    if (TRANSB == 0) {
      const float* bp = Bb + (long)(k0 + brow) * ldb + (n0 + bcol);
      async_copy_b128(lds_off_of(&Bs[buf][brow][bcol]), bp);
    } else {   // B used as B^T: Bs[kk][n] = B[n][kk] (strided LDS dest)
      const float* bp = Bb + (long)(n0 + tn) * ldb + (k0 + tkk);
      async_copy_b32(lds_off_of(&Bs[buf][tkk + 0][tn]), bp + 0);
      async_copy_b32(lds_off_of(&Bs[buf][tkk + 1][tn]), bp + 1);
      async_copy_b32(lds_off_of(&Bs[buf][tkk + 2][tn]), bp + 2);
      async_copy_b32(lds_off_of(&Bs[buf][tkk + 3][tn]), bp + 3);
    }
  };

  v8f acc00 = {}, acc01 = {}, acc10 = {}, acc11 = {};

  int nk = K >> 4;
  issue_tile(0, 0);
  for (int t = 0; t < nk; ++t) {
    int cur = t & 1;
    if (t + 1 < nk) {
      issue_tile((t + 1) << 4, cur ^ 1);
      wait_asynccnt<NPF>();    // current tile done; next prefetch in flight
    } else {
      wait_asynccnt<0>();
    }
    __syncthreads();

    const float (*Ac)[APAD] = As[cur];
    const float (*Bc)[BPAD] = Bs[cur];
#pragma unroll
    for (int kk = 0; kk < 16; kk += 4) {
      // f32 A 16x4 fragment: lanes 0-15 hold K=kk,kk+1; lanes 16-31 hold K=kk+2,kk+3
      int krow = kk + sel * 2;
      v2f a0, a1, b0, b1;
      a0.x = Ac[wm * 32 + lrow][krow];
      a0.y = Ac[wm * 32 + lrow][krow + 1];
      a1.x = Ac[wm * 32 + 16 + lrow][krow];
      a1.y = Ac[wm * 32 + 16 + lrow][krow + 1];
      // f32 B 4x16 fragment: rows striped across lanes; VGPR0 rows K,K+2; VGPR1 rows K+1,K+3
      b0.x = Bc[krow][wn * 32 + lrow];
      b0.y = Bc[krow + 1][wn * 32 + lrow];
      b1.x = Bc[krow][wn * 32 + 16 + lrow];
      b1.y = Bc[krow + 1][wn * 32 + 16 + lrow];
      acc00 = __builtin_amdgcn_wmma_f32_16x16x4_f32(false, a0, false, b0,
                                                    (short)0, acc00, false, false);
      acc01 = __builtin_amdgcn_wmma_f32_16x16x4_f32(false, a0, false, b1,
                                                    (short)0, acc01, false, false);
      acc10 = __builtin_amdgcn_wmma_f32_16x16x4_f32(false, a1, false, b0,
                                                    (short)0, acc10, false, false);
      acc11 = __builtin_amdgcn_wmma_f32_16x16x4_f32(false, a1, false, b1,
                                                    (short)0, acc11, false, false);
    }
    __syncthreads();
  }

  // C/D layout: VGPR r, lanes 0-15 -> M=r; lanes 16-31 -> M=r+8
  int c0 = n0 + wn * 32 + lrow;
  int c1 = c0 + 16;
  int rowBase = m0 + wm * 32 + sel * 8;
#pragma unroll
  for (int r = 0; r < 8; r++) {
    int rowA = rowBase + r;        // acc00 / acc01
    int rowB = rowBase + 16 + r;   // acc10 / acc11
    float v00 = acc00[r] * alpha, v01 = acc01[r] * alpha;
    float v10 = acc10[r] * alpha, v11 = acc11[r] * alpha;
    if (bias) {
      v00 += bias[c0]; v01 += bias[c1];
      v10 += bias[c0]; v11 += bias[c1];
    }
    if (act_gelu) {
      v00 = gelu_f(v00); v01 = gelu_f(v01);
      v10 = gelu_f(v10); v11 = gelu_f(v11);
    }
    if (residual) {
      v00 += residual[(long)rowA * ldc + c0]; v01 += residual[(long)rowA * ldc + c1];
      v10 += residual[(long)rowB * ldc + c0]; v11 += residual[(long)rowB * ldc + c1];
    }
    Cb[(long)rowA * ldc + c0] = v00;
    Cb[(long)rowA * ldc + c1] = v01;
    Cb[(long)rowB * ldc + c0] = v10;
    Cb[(long)rowB * ldc + c1] = v11;
  }
}

// ---------------------------------------------------------------- qkv permute
// qkv [B,C,3D] -> q/k/v [B,H,C,HD]
__global__ void qkv_permute_kernel(const float* __restrict__ qkv,
                                   float* __restrict__ q, float* __restrict__ k,
                                   float* __restrict__ v) {
  long idx = (long)blockIdx.x * blockDim.x + threadIdx.x;   // over B*C*D
  if (idx >= (long)NT * D) return;
  int  d = idx % HD;
  long t2 = idx / HD;
  int  h = t2 % H;
  long t = t2 / H;                 // token = b*CTX + c
  long b = t / CTX;
  long c = t % CTX;
  long dst = (((long)b * H + h) * CTX + c) * HD + d;
  const float* src = qkv + t * (3 * D);
  q[dst] = src[0 * D + h * HD + d];
  k[dst] = src[1 * D + h * HD + d];
  v[dst] = src[2 * D + h * HD + d];
}

// ---------------------------------------------------------------- masked softmax
// Faithful to reference: logits MULTIPLIED by tril(k=-1) mask, i.e.
// l_j = s_j for j < q, else 0; softmax over all CTX positions.
__global__ __launch_bounds__(256)
void softmax_mask_kernel(float* __restrict__ scores) {
  __shared__ float red[256];
  long rowid = blockIdx.x;          // over B*H*CTX
  int  qpos  = (int)(rowid % CTX);
  float* row = scores + rowid * (long)CTX;

  float mx = -INFINITY;
  for (int j = threadIdx.x; j < CTX; j += 256) {
    float lv = (j < qpos) ? row[j] : 0.0f;
    mx = fmaxf(mx, lv);
  }
  red[threadIdx.x] = mx; __syncthreads();
  for (int o = 128; o > 0; o >>= 1) {
    if (threadIdx.x < o) red[threadIdx.x] = fmaxf(red[threadIdx.x], red[threadIdx.x + o]);
    __syncthreads();
  }
  mx = red[0]; __syncthreads();

  float sum = 0.f;
  for (int j = threadIdx.x; j < CTX; j += 256) {
    float lv = (j < qpos) ? row[j] : 0.0f;
    float e = expf(lv - mx);
    row[j] = e;
    sum += e;
  }
  red[threadIdx.x] = sum; __syncthreads();
  for (int o = 128; o > 0; o >>= 1) {
    if (threadIdx.x < o) red[threadIdx.x] += red[threadIdx.x + o];
    __syncthreads();
  }
  float inv = 1.0f / red[0];
  for (int j = threadIdx.x; j < CTX; j += 256)
    row[j] *= inv;
}

// ---------------------------------------------------------------- host driver
extern "C" void kernel_launch(void* const* d_in, const int* in_sizes, int n_in,
                              void* d_out, int out_size, void* d_ws, size_t ws_size,
                              hipStream_t stream) {
  (void)in_sizes; (void)n_in; (void)out_size; (void)ws_size;
  const int*   x       = (const int*)d_in[0];
  const float* tok_emb = (const float*)d_in[1];
  const float* pos_emb = (const float*)d_in[2];
  const float* ln1_w   = (const float*)d_in[3];
  const float* ln1_b   = (const float*)d_in[4];
  const float* att_w   = (const float*)d_in[5];
  const float* att_b   = (const float*)d_in[6];
  const float* rp_w    = (const float*)d_in[7];
  const float* rp_b    = (const float*)d_in[8];
  const float* ln2_w   = (const float*)d_in[9];
  const float* ln2_b   = (const float*)d_in[10];
  const float* fc1_w   = (const float*)d_in[11];
  const float* fc1_b   = (const float*)d_in[12];
  const float* fc2_w   = (const float*)d_in[13];
  const float* fc2_b   = (const float*)d_in[14];
  const float* lnf_w   = (const float*)d_in[15];
  const float* lnf_b   = (const float*)d_in[16];
  const float* head_w  = (const float*)d_in[17];
  const float* head_b  = (const float*)d_in[18];

  const long SZ = (long)NT * D;        // 2,097,152 floats
  float* ws      = (float*)d_ws;
  float* h       = ws + 0 * SZ;
  float* hn      = ws + 1 * SZ;
  float* hn2     = ws + 2 * SZ;
  float* qkv     = ws + 3 * SZ;        // 3*SZ
  float* attnout = qkv;                // alias: qkv dead after permute
  float* q       = ws + 6 * SZ;
  float* k       = ws + 7 * SZ;
  float* v       = ws + 8 * SZ;
  float* scores  = ws + 9 * SZ;        // 16*SZ  (B*H*CTX*CTX)
  float* ffn1    = scores;             // alias: scores dead after attn

  auto gemm = [&](const float* A, const float* Bm, const float* bias, const float* res,
                  float* Cm, int M, int N, int K, int lda, int ldb, int ldc,
                  long sA, long sB, int gC, long sC1, long sC2,
                  float alpha, int transB, int gelu, int batch) {
    dim3 grid(N / 64, M / 128, batch);
    if (transB)
      gemm_wmma_kernel<1><<<grid, 256, 0, stream>>>(A, Bm, bias, res, Cm,
                                                    M, N, K, lda, ldb, ldc,
                                                    sA, sB, gC, sC1, sC2, alpha, gelu);
    else
      gemm_wmma_kernel<0><<<grid, 256, 0, stream>>>(A, Bm, bias, res, Cm,
                                                    M, N, K, lda, ldb, ldc,
                                                    sA, sB, gC, sC1, sC2, alpha, gelu);
  };

  embed_kernel<<<NT, 256, 0, stream>>>(x, tok_emb, pos_emb, h);

  const float scale = 0.125f;          // 1/sqrt(HD)
  for (int i = 0; i < L; ++i) {
    // hn = ln1(h)
    ln_kernel<<<NT, 256, 0, stream>>>(h, ln1_w + (long)i * D, ln1_b + (long)i * D, hn);
    // qkv = hn @ att_w[i] + att_b[i]
    gemm(hn, att_w + (long)i * D * 3 * D, att_b + (long)i * 3 * D, nullptr, qkv,
         NT, 3 * D, D, D, 3 * D, 3 * D, 0, 0, 1, 0, 0, 1.0f, 0, 0, 1);
    // split/permute to [B,H,C,HD]
    {
      long n = (long)NT * D;
      qkv_permute_kernel<<<(unsigned)((n + 255) / 256), 256, 0, stream>>>(qkv, q, k, v);
    }
    // scores = scale * q @ k^T   (batched over B*H)
    gemm(q, k, nullptr, nullptr, scores,
         CTX, CTX, HD, HD, HD, CTX,
         (long)CTX * HD, (long)CTX * HD, 1, 0, (long)CTX * CTX,
         scale, 1, 0, BB * H);
    // multiplicative-mask softmax
    softmax_mask_kernel<<<BB * H * CTX, 256, 0, stream>>>(scores);
    // attnout[b, c, h*HD + d] = P @ v   (batched, strided un-permuting store)
    gemm(scores, v, nullptr, nullptr, attnout,
         CTX, HD, CTX, CTX, HD, D,
         (long)CTX * CTX, (long)CTX * HD, H, (long)HD, (long)CTX * D,
         1.0f, 0, 0, BB * H);
    // h = hn + (attnout @ rp_w[i] + rp_b[i])   -- residual adds POST-LN act
    gemm(attnout, rp_w + (long)i * D * D, rp_b + (long)i * D, hn, h,
         NT, D, D, D, D, D, 0, 0, 1, 0, 0, 1.0f, 0, 0, 1);
    // hn2 = ln2(h)
    ln_kernel<<<NT, 256, 0, stream>>>(h, ln2_w + (long)i * D, ln2_b + (long)i * D, hn2);
    // ffn1 = gelu(hn2 @ fc1_w[i] + fc1_b[i])
    gemm(hn2, fc1_w + (long)i * D * 4 * D, fc1_b + (long)i * 4 * D, nullptr, ffn1,
         NT, 4 * D, D, D, 4 * D, 4 * D, 0, 0, 1, 0, 0, 1.0f, 0, 1, 1);
    // h = hn2 + (ffn1 @ fc2_w[i] + fc2_b[i])
    gemm(ffn1, fc2_w + (long)i * 4 * D * D, fc2_b + (long)i * D, hn2, h,
         NT, D, 4 * D, 4 * D, D, D, 0, 0, 1, 0, 0, 1.0f, 0, 0, 1);
  }

  // final LN + LM head
  ln_kernel<<<NT, 256, 0, stream>>>(h, lnf_w, lnf_b, hn);
  gemm(hn, head_w, head_b, nullptr, (float*)d_out,
       NT, V, D, D, V, V, 0, 0, 1, 0, 0, 1.0f, 0, 0, 1);
}